// WindowAttention_85341000171895
// MI455X (gfx1250) — compile-verified
//
#include <hip/hip_runtime.h>

typedef __attribute__((ext_vector_type(2))) float v2f;
typedef __attribute__((ext_vector_type(8))) float v8f;

#define DIM       128
#define NTOK      49
#define NPAD      64
#define HEADS     4
#define HD        32
#define NWIN      1024
#define THREE_DIM 384
#define SCALE     0.17677669529663687f   // 1/sqrt(32)

// LDS strides padded so stride % 64 == 4 -> conflict-free column access
#define X_STRIDE   132
#define QKV_STRIDE 388
#define S_STRIDE   68

#define SM_QKV   (NPAD * X_STRIDE)                      // 8448 floats
#define SM_S     (SM_QKV + NPAD * QKV_STRIDE)           // 33280 floats
#define SM_TOTAL (SM_S + HEADS * NPAD * S_STRIDE)       // 50688 floats = 202752 B

// Generic 16x4 fragment loader matching CDNA5 WMMA f32 VGPR layout:
// lane&15 indexes the 16-wide dim (M for A, N for B); lane>>4 selects K pair.
__device__ __forceinline__ v2f ld_frag(const float* __restrict__ p,
                                       int s16, int sk, int i16, int k0, int lane) {
    const int r    = lane & 15;
    const int half = lane >> 4;
    const long base = (long)(i16 + r) * s16 + (long)(k0 + 2 * half) * sk;
    v2f v;
    v.x = p[base];
    v.y = p[base + sk];
    return v;
}

__global__ __launch_bounds__(256) void swin_window_attn_fused(
    const float* __restrict__ x,        // (B,49,128)
    const float* __restrict__ mask,     // (1024,49,49)
    const float* __restrict__ qkv_w,    // (128,384)
    const float* __restrict__ qkv_b,    // (384)
    const float* __restrict__ proj_w,   // (128,128)
    const float* __restrict__ proj_b,   // (128)
    const float* __restrict__ bias_table, // (169,4)
    const int*   __restrict__ rel_idx,  // (49,49)
    float* __restrict__ out)            // (B,49,128)
{
    extern __shared__ float smem[];
    float* xbuf = smem;            // 64 x 128 (stride 132): x, later attn-out
    float* qkvs = smem + SM_QKV;   // 64 x 384 (stride 388): q|k|v
    float* sbuf = smem + SM_S;     // 4 x 64 x 64 (stride 68): scores/probs

    const int w    = blockIdx.x;           // window id
    const int tid  = threadIdx.x;
    const int lane = tid & 31;
    const int wave = tid >> 5;             // 0..7 (wave32)
    const int r16  = lane & 15;
    const int half = lane >> 4;

    // ---- Phase 0: stage x window into LDS, zero-pad rows 49..63 ----
    for (int i = tid; i < NPAD * DIM; i += 256) {
        const int row = i >> 7, col = i & 127;
        float v = 0.f;
        if (row < NTOK) v = x[((long)w * NTOK + row) * DIM + col];
        xbuf[row * X_STRIDE + col] = v;
    }
    __syncthreads();

    // ---- Phase 1: QKV GEMM  (64x384) = xbuf(64x128) @ qkv_w(128x384) + b ----
    // q pre-scaled by 1/sqrt(hd) on store.
    for (int t = wave; t < 4 * 24; t += 8) {
        const int m0 = (t / 24) * 16;
        const int n0 = (t % 24) * 16;
        v8f acc = {};
        for (int k = 0; k < DIM; k += 4) {
            v2f a = ld_frag(xbuf,  X_STRIDE, 1,         m0, k, lane);
            v2f b = ld_frag(qkv_w, 1,        THREE_DIM, n0, k, lane);
            acc = __builtin_amdgcn_wmma_f32_16x16x4_f32(
                false, a, false, b, (short)0, acc, false, false);
        }
        const int col = n0 + r16;
        const float bias = qkv_b[col];
        const float s = (col < DIM) ? SCALE : 1.0f;   // scale only q columns
        for (int r = 0; r < 8; ++r) {
            const int row = m0 + half * 8 + r;
            qkvs[row * QKV_STRIDE + col] = (acc[r] + bias) * s;
        }
    }
    __syncthreads();

    // ---- Phase 2: scores  S[h] (64x64) = q_h(64x32) @ k_h^T(32x64) + rpb + mask
    const int mw = w % NWIN;
    for (int t = wave; t < HEADS * 4 * 4; t += 8) {
        const int h  = t >> 4;
        const int m0 = ((t >> 2) & 3) * 16;
        const int n0 = (t & 3) * 16;
        const float* qp = qkvs + h * HD;          // q columns for head h
        const float* kp = qkvs + DIM + h * HD;    // k columns for head h
        v8f acc = {};
        for (int k = 0; k < HD; k += 4) {
            v2f a = ld_frag(qp, QKV_STRIDE, 1, m0, k, lane);
            v2f b = ld_frag(kp, QKV_STRIDE, 1, n0, k, lane);  // B = K^T
            acc = __builtin_amdgcn_wmma_f32_16x16x4_f32(
                false, a, false, b, (short)0, acc, false, false);
        }
        const int key = n0 + r16;
        float* srow_base = sbuf + h * NPAD * S_STRIDE;
        for (int r = 0; r < 8; ++r) {
            const int row = m0 + half * 8 + r;    // query index
            float val;
            if (key < NTOK) {
                val = acc[r];
                if (row < NTOK) {
                    val += bias_table[rel_idx[row * NTOK + key] * HEADS + h];
                    val += mask[((long)mw * NTOK + row) * NTOK + key];
                }
            } else {
                val = -1.0e30f;                   // padded keys excluded
            }
            srow_base[row * S_STRIDE + key] = val;
        }
    }
    __syncthreads();

    // ---- Phase 3: softmax over keys (49 valid). 256 threads = 4 heads x 64 rows
    {
        const int h = tid >> 6, row = tid & 63;
        float* srow = sbuf + (h * NPAD + row) * S_STRIDE;
        float m = -1.0e30f;
        for (int k = 0; k < NTOK; ++k) m = fmaxf(m, srow[k]);
        float sum = 0.f;
        for (int k = 0; k < NTOK; ++k) {
            const float e = __expf(srow[k] - m);
            srow[k] = e;
            sum += e;
        }
        const float inv = 1.0f / sum;
        for (int k = 0; k < NTOK; ++k) srow[k] *= inv;
        for (int k = NTOK; k < NPAD; ++k) srow[k] = 0.f;
    }
    __syncthreads();

    // ---- Phase 4: O_h (64x32) = P_h(64x64) @ V_h(64x32), into xbuf (reuse) ----
    for (int t = wave; t < HEADS * 4 * 2; t += 8) {
        const int h  = t >> 3;
        const int m0 = ((t >> 1) & 3) * 16;
        const int n0 = (t & 1) * 16;
        const float* pp = sbuf + h * NPAD * S_STRIDE;
        const float* vp = qkvs + 2 * DIM + h * HD;
        v8f acc = {};
        for (int k = 0; k < NPAD; k += 4) {
            v2f a = ld_frag(pp, S_STRIDE, 1,          m0, k, lane);
            v2f b = ld_frag(vp, 1,        QKV_STRIDE, n0, k, lane);  // V row-major
            acc = __builtin_amdgcn_wmma_f32_16x16x4_f32(
                false, a, false, b, (short)0, acc, false, false);
        }
        const int col = h * HD + n0 + r16;
        for (int r = 0; r < 8; ++r) {
            const int row = m0 + half * 8 + r;
            xbuf[row * X_STRIDE + col] = acc[r];
        }
    }
    __syncthreads();

    // ---- Phase 5: final proj (64x128) = O(64x128) @ proj_w(128x128) + b ----
    for (int t = wave; t < 4 * 8; t += 8) {
        const int m0 = (t >> 3) * 16;
        const int n0 = (t & 7) * 16;
        v8f acc = {};
        for (int k = 0; k < DIM; k += 4) {
            v2f a = ld_frag(xbuf,   X_STRIDE, 1,   m0, k, lane);
            v2f b = ld_frag(proj_w, 1,        DIM, n0, k, lane);
            acc = __builtin_amdgcn_wmma_f32_16x16x4_f32(
                false, a, false, b, (short)0, acc, false, false);
        }
        const int col = n0 + r16;
        const float bias = proj_b[col];
        for (int r = 0; r < 8; ++r) {
            const int row = m0 + half * 8 + r;
            if (row < NTOK)
                out[((long)w * NTOK + row) * DIM + col] = acc[r] + bias;
        }
    }
}

extern "C" void kernel_launch(void* const* d_in, const int* in_sizes, int n_in,
                              void* d_out, int out_size, void* d_ws, size_t ws_size,
                              hipStream_t stream) {
    const float* x          = (const float*)d_in[0];
    const float* mask       = (const float*)d_in[1];
    const float* qkv_w      = (const float*)d_in[2];
    const float* qkv_b      = (const float*)d_in[3];
    const float* proj_w     = (const float*)d_in[4];
    const float* proj_b     = (const float*)d_in[5];
    const float* bias_table = (const float*)d_in[6];
    const int*   rel_idx    = (const int*)d_in[7];
    float* out = (float*)d_out;

    const int n_windows = in_sizes[0] / (NTOK * DIM);   // 8192
    const size_t smem_bytes = (size_t)SM_TOTAL * sizeof(float);  // ~203 KB of 320 KB WGP LDS

    hipFuncSetAttribute((const void*)swin_window_attn_fused,
                        hipFuncAttributeMaxDynamicSharedMemorySize,
                        (int)smem_bytes);

    swin_window_attn_fused<<<n_windows, 256, smem_bytes, stream>>>(
        x, mask, qkv_w, qkv_b, proj_w, proj_b, bias_table, rel_idx, out);
}